// VGG16RoIHead_7808250544641
// MI455X (gfx1250) — compile-verified
//
#include <hip/hip_runtime.h>
#include <hip/hip_bf16.h>
#include <math.h>

typedef __attribute__((ext_vector_type(2))) float v2f;
typedef __attribute__((ext_vector_type(8))) float v8f;

#define RROI   128
#define CCH    512
#define HH     32
#define WW     32
#define KFLAT  25088   /* 512 * 7 * 7 */
#define NFC    4096
#define NLOC   84      /* 21 classes * 4 */
#define NSC    21

#define KT       64            /* K-tile staged in LDS per buffer          */
#define LDSTRIDE 68            /* padded row stride (floats): 16B aligned, */
                               /* banks (4*row + k) % 64 -> conflict-free  */
#define BUFFLOATS (128 * LDSTRIDE)
#define BUFBYTES  (BUFFLOATS * 4)

/* ---------------------------------------------------------------------- */
/* 1) torchvision-style quantized RoIPool -> flat [128, 25088]            */
/* ---------------------------------------------------------------------- */
__global__ void roipool_flat_kernel(const float* __restrict__ x,
                                    const float* __restrict__ rois,
                                    const int*   __restrict__ roi_idx,
                                    float*       __restrict__ Aflat) {
    int gid = blockIdx.x * blockDim.x + threadIdx.x;
    const int total = RROI * KFLAT;
    if (gid >= total) return;

    int r   = gid / KFLAT;
    int k   = gid - r * KFLAT;
    int c   = k / 49;
    int bin = k - c * 49;
    int ph  = bin / 7;
    int pw  = bin - ph * 7;

    float y1 = rois[r * 4 + 0];
    float x1 = rois[r * 4 + 1];
    float y2 = rois[r * 4 + 2];
    float x2 = rois[r * 4 + 3];

    const float S = 1.0f / 16.0f;
    float sw = rintf(x1 * S), sh = rintf(y1 * S);   /* rint == jnp.round */
    float ew = rintf(x2 * S), eh = rintf(y2 * S);
    float bw = fmaxf(ew - sw + 1.0f, 1.0f) * (1.0f / 7.0f);
    float bh = fmaxf(eh - sh + 1.0f, 1.0f) * (1.0f / 7.0f);

    float wsf = fminf(fmaxf(floorf((float)pw * bw) + sw, 0.0f), (float)WW);
    float wef = fminf(fmaxf(ceilf((float)(pw + 1) * bw) + sw, 0.0f), (float)WW);
    float hsf = fminf(fmaxf(floorf((float)ph * bh) + sh, 0.0f), (float)HH);
    float hef = fminf(fmaxf(ceilf((float)(ph + 1) * bh) + sh, 0.0f), (float)HH);

    int ws = (int)wsf, we = (int)wef, hs = (int)hsf, he = (int)hef;
    int n = roi_idx[r];

    const float* p = x + ((size_t)n * CCH + (size_t)c) * (HH * WW);
    float m = -INFINITY;
    for (int h = hs; h < he; ++h)
        for (int w = ws; w < we; ++w)
            m = fmaxf(m, p[h * WW + w]);
    if (hs >= he || ws >= we) m = 0.0f;
    Aflat[gid] = m;
}

/* ---------------------------------------------------------------------- */
/* 2) C[r][j] = bias[j]                                                   */
/* ---------------------------------------------------------------------- */
__global__ void bias_init_kernel(float* __restrict__ C,
                                 const float* __restrict__ bias,
                                 int ncols, int total) {
    int gid = blockIdx.x * blockDim.x + threadIdx.x;
    if (gid >= total) return;
    C[gid] = bias[gid % ncols];
}

/* ---------------------------------------------------------------------- */
/* Issue one 128x64 fp32 A-tile (this thread's 8 float4 slices) as CDNA5  */
/* async global->LDS copies. Tracked on ASYNCcnt; in-order completion.    */
/* ---------------------------------------------------------------------- */
__device__ __forceinline__ void stage_tile_async(unsigned lds_byte,
                                                 const float* g) {
#pragma unroll
    for (int i = 0; i < 8; ++i) {
        asm volatile("global_load_async_to_lds_b128 %0, %1, off"
                     :: "v"(lds_byte + 16u * (unsigned)i), "v"(g + 4 * i)
                     : "memory");
    }
}

/* ---------------------------------------------------------------------- */
/* 3) fp32 WMMA GEMM partial: C[128,N] += A[128,K] * W[K,N] over a K-chunk */
/*    256 threads = 8 wave32; wave w owns 16 N columns, 8 accumulators     */
/*    cover all M=128 rows. A tiles double-buffered in LDS via async DMA;  */
/*    fragments software-pipelined one K-step ahead of the WMMAs.          */
/* ---------------------------------------------------------------------- */
__global__ void __launch_bounds__(256)
gemm_f32_wmma_kernel(const float* __restrict__ A,
                     const float* __restrict__ Wm,
                     float*       __restrict__ C,
                     int K, int Kc, int N) {
    __shared__ __align__(16) float smem[2 * BUFFLOATS];

    const int tid   = threadIdx.x;
    const int lane  = tid & 31;
    const int wave  = tid >> 5;
    const int mrow  = lane & 15;
    const int khalf = (lane >> 4) * 2;          /* K sub-pair of half-wave */
    const int ncol  = blockIdx.x * 128 + wave * 16 + mrow;
    const int kbeg  = blockIdx.y * Kc;
    int kend = kbeg + Kc; if (kend > K) kend = K;

    /* stager: thread -> (row = tid>>1, half = tid&1), 8 x b128 each      */
    const int srow  = tid >> 1;
    const int shalf = tid & 1;
    const unsigned smem_base = (unsigned)(unsigned long long)(&smem[0]);
    const unsigned stage_lds = smem_base
                             + (unsigned)(srow * LDSTRIDE + shalf * 32) * 4u;
    const float* gstage = A + (size_t)srow * K + kbeg + shalf * 32;

    v8f zero = {0.f, 0.f, 0.f, 0.f, 0.f, 0.f, 0.f, 0.f};
    v8f acc[8];
#pragma unroll
    for (int t = 0; t < 8; ++t) acc[t] = zero;

    /* W stream: two incremented pointers, k rows khalf / khalf+1         */
    const float* bp0 = Wm + (size_t)(kbeg + khalf) * N + ncol;
    const float* bp1 = bp0 + N;

    /* prologue: stage tile 0 into buffer 0 */
    stage_tile_async(stage_lds, gstage);

    int buf = 0;
    for (int kt = kbeg; kt < kend; kt += KT) {
        const bool has_next = (kt + KT) < kend;
        if (has_next) {
            stage_tile_async(stage_lds + (unsigned)(buf ^ 1) * (unsigned)BUFBYTES,
                             gstage + (kt - kbeg) + KT);
            asm volatile("s_wait_asynccnt 0x8" ::: "memory"); /* cur tile done */
        } else {
            asm volatile("s_wait_asynccnt 0x0" ::: "memory");
        }
        __syncthreads();                    /* tile visible to all waves   */

        const float* sa0 = smem + buf * BUFFLOATS + mrow * LDSTRIDE + khalf;

        /* ---- software-pipelined tile compute: load step k+1, mma step k */
        v2f bcur, acur[8];
        bcur.x = bp0[0];
        bcur.y = bp1[0];
        bp0 += (size_t)4 * N;
        bp1 += (size_t)4 * N;
#pragma unroll
        for (int t = 0; t < 8; ++t)
            acur[t] = *(const v2f*)(sa0 + t * (16 * LDSTRIDE));

#pragma unroll
        for (int kk = 0; kk < KT; kk += 4) {
            v2f bnxt, anxt[8];
            if (kk + 4 < KT) {              /* compile-time after unroll   */
                __builtin_prefetch(bp0 + (size_t)64 * N, 0, 1);
                bnxt.x = bp0[0];
                bnxt.y = bp1[0];
                bp0 += (size_t)4 * N;
                bp1 += (size_t)4 * N;
                const float* sa = sa0 + (kk + 4);
#pragma unroll
                for (int t = 0; t < 8; ++t)
                    anxt[t] = *(const v2f*)(sa + t * (16 * LDSTRIDE));
            }
#pragma unroll
            for (int t = 0; t < 8; ++t) {
                acc[t] = __builtin_amdgcn_wmma_f32_16x16x4_f32(
                             false, acur[t], false, bcur, (short)0, acc[t],
                             false, false);
            }
            if (kk + 4 < KT) {
                bcur = bnxt;
#pragma unroll
                for (int t = 0; t < 8; ++t) acur[t] = anxt[t];
            }
        }
        __syncthreads();                    /* done reading buf -> restageable */
        buf ^= 1;
    }

    /* epilogue: one base pointer, compile-time row offsets fold into imm */
    const int rsel = (lane >> 4) * 8;
    float* cp = C + (size_t)rsel * N + ncol;
#pragma unroll
    for (int t = 0; t < 8; ++t) {
#pragma unroll
        for (int e = 0; e < 8; ++e) {
            atomicAdd(cp + (size_t)(t * 16 + e) * N, acc[t][e]);
        }
    }
}

/* ---------------------------------------------------------------------- */
/* 4) in-place ReLU                                                       */
/* ---------------------------------------------------------------------- */
__global__ void relu_kernel(float* __restrict__ C, int total) {
    int gid = blockIdx.x * blockDim.x + threadIdx.x;
    if (gid >= total) return;
    C[gid] = fmaxf(C[gid], 0.0f);
}

/* ---------------------------------------------------------------------- */
/* 5) heads: out = [fc7 @ W_loc + b_loc | fc7 @ W_score + b_score]        */
/* ---------------------------------------------------------------------- */
__global__ void heads_kernel(const float* __restrict__ fc7,
                             const float* __restrict__ Wloc,
                             const float* __restrict__ bloc,
                             const float* __restrict__ Wsc,
                             const float* __restrict__ bsc,
                             float* __restrict__ out) {
    int gid = blockIdx.x * blockDim.x + threadIdx.x;
    const int NL = RROI * NLOC;
    const int NS = RROI * NSC;
    if (gid >= NL + NS) return;

    const float* Wp; const float* bp; int r, j, ncols;
    if (gid < NL) { r = gid / NLOC; j = gid % NLOC; Wp = Wloc; bp = bloc; ncols = NLOC; }
    else { int g = gid - NL; r = g / NSC; j = g % NSC; Wp = Wsc; bp = bsc; ncols = NSC; }

    float s = bp[j];
    const float* f = fc7 + (size_t)r * NFC;
    for (int k = 0; k < NFC; ++k)
        s = fmaf(f[k], Wp[(size_t)k * ncols + j], s);
    out[gid] = s;
}

/* ---------------------------------------------------------------------- */
extern "C" void kernel_launch(void* const* d_in, const int* in_sizes, int n_in,
                              void* d_out, int out_size, void* d_ws, size_t ws_size,
                              hipStream_t stream) {
    (void)in_sizes; (void)n_in; (void)out_size; (void)ws_size;

    const float* x       = (const float*)d_in[0];
    const float* rois    = (const float*)d_in[1];
    const int*   roi_idx = (const int*)  d_in[2];
    const float* W1      = (const float*)d_in[3];
    const float* b1      = (const float*)d_in[4];
    const float* W2      = (const float*)d_in[5];
    const float* b2      = (const float*)d_in[6];
    const float* W_loc   = (const float*)d_in[7];
    const float* b_loc   = (const float*)d_in[8];
    const float* W_score = (const float*)d_in[9];
    const float* b_score = (const float*)d_in[10];
    float* out = (float*)d_out;

    /* workspace layout (floats): Aflat | fc6 | fc7  (~16.3 MB total)     */
    float* Aflat = (float*)d_ws;                     /* 128*25088 */
    float* fc6   = Aflat + (size_t)RROI * KFLAT;     /* 128*4096  */
    float* fc7   = fc6   + (size_t)RROI * NFC;       /* 128*4096  */

    const int BLK = 256;

    /* 1) RoIPool -> flat */
    {
        int total = RROI * KFLAT;
        roipool_flat_kernel<<<(total + BLK - 1) / BLK, BLK, 0, stream>>>(
            x, rois, roi_idx, Aflat);
    }

    /* 2) fc6 = relu(Aflat @ W1 + b1) : 8 K-chunks of 3136 (49 LDS tiles) */
    {
        int total = RROI * NFC;
        bias_init_kernel<<<(total + BLK - 1) / BLK, BLK, 0, stream>>>(fc6, b1, NFC, total);
        dim3 grid(NFC / 128, 8);
        gemm_f32_wmma_kernel<<<grid, BLK, 0, stream>>>(Aflat, W1, fc6,
                                                       KFLAT, KFLAT / 8, NFC);
        relu_kernel<<<(total + BLK - 1) / BLK, BLK, 0, stream>>>(fc6, total);
    }

    /* 3) fc7 = relu(fc6 @ W2 + b2) : 4 K-chunks of 1024 (16 LDS tiles)   */
    {
        int total = RROI * NFC;
        bias_init_kernel<<<(total + BLK - 1) / BLK, BLK, 0, stream>>>(fc7, b2, NFC, total);
        dim3 grid(NFC / 128, 4);
        gemm_f32_wmma_kernel<<<grid, BLK, 0, stream>>>(fc6, W2, fc7,
                                                       NFC, NFC / 4, NFC);
        relu_kernel<<<(total + BLK - 1) / BLK, BLK, 0, stream>>>(fc7, total);
    }

    /* 4) heads -> d_out */
    {
        int total = RROI * (NLOC + NSC);
        heads_kernel<<<(total + BLK - 1) / BLK, BLK, 0, stream>>>(
            fc7, W_loc, b_loc, W_score, b_score, out);
    }
}